// GATout_29214367547581
// MI455X (gfx1250) — compile-verified
//
#include <hip/hip_runtime.h>
#include <hip/hip_bf16.h>

#define N_SRC 100000
#define N_DST 20000
#define E_CNT 1250000
#define D 64
#define NEG_SLOPE 0.2f

typedef __attribute__((ext_vector_type(2))) float v2f;
typedef __attribute__((ext_vector_type(8))) float v8f;

// ---------------------------------------------------------------------------
// Kernel 0: init out = bias (broadcast), denom = 0, m_enc = 0 (== encoded -inf)
// ---------------------------------------------------------------------------
__global__ __launch_bounds__(256) void gat_init(float* __restrict__ out,
                                                const float* __restrict__ bias,
                                                float* __restrict__ denom,
                                                unsigned* __restrict__ m_enc) {
    int i = blockIdx.x * blockDim.x + threadIdx.x;
    if (i < N_DST * D) out[i] = bias[i & (D - 1)];
    if (i < N_DST) {
        denom[i] = 0.0f;
        m_enc[i] = 0u;  // below every encoded finite float
    }
}

// ---------------------------------------------------------------------------
// Kernel 1: watt[0:64] = W @ att_src ; watt[64:128] = W @ att_dst
// ---------------------------------------------------------------------------
__global__ __launch_bounds__(64) void gat_watt(const float* __restrict__ W,
                                               const float* __restrict__ att_src,
                                               const float* __restrict__ att_dst,
                                               float* __restrict__ watt) {
    int t = threadIdx.x;  // 0..63 = row of W
    float s = 0.0f, d = 0.0f;
    #pragma unroll 8
    for (int j = 0; j < D; ++j) {
        float w = W[t * D + j];
        s += w * att_src[j];
        d += w * att_dst[j];
    }
    watt[t] = s;
    watt[D + t] = d;
}

// ---------------------------------------------------------------------------
// Kernel 2: h_src = x @ W  via V_WMMA_F32_16X16X4_F32 (fp32-exact).
// Block = 128 threads = 4 waves; each wave computes a 16x64 output tile.
// W (64x64 f32, 16 KB) staged in LDS once per block.
//
// Branch-free A-fragment handling: OOB rows are clamped to row N_SRC-1 and the
// fragment scaled by 0.0f, so EXEC stays all-ones (required by WMMA) and all
// 16 global_load_b64's issue as one clause before the WMMA stream.
//
// ISA layouts (CDNA5 ISA §7.12.2):
//   A 16x4 f32 : lanes 0-15 -> M=lane, v0=K0 v1=K1 ; lanes 16-31 -> v0=K2 v1=K3
//   B row-striped across lanes within a VGPR (v0: K0/K2 halves, v1: K1/K3)
//   C/D 16x16 f32: VGPR r -> row r (lanes 0-15), row r+8 (lanes 16-31)
// ---------------------------------------------------------------------------
__global__ __launch_bounds__(128) void gat_gemm_hsrc(const float* __restrict__ x,
                                                     const float* __restrict__ W,
                                                     float* __restrict__ h) {
    __shared__ float ldsW[D * D];
    for (int i = threadIdx.x; i < D * D; i += 128) ldsW[i] = W[i];
    __syncthreads();

    const int lane = threadIdx.x & 31;
    const int mrow = lane & 15;          // M (and N) sub-index
    const int hi   = lane >> 4;          // 0: K pair {0,1}; 1: K pair {2,3}
    const int row0 = blockIdx.x * 64 + (threadIdx.x >> 5) * 16;
    const int row  = row0 + mrow;
    const int rowc = (row < N_SRC) ? row : (N_SRC - 1);
    const float valid = (row < N_SRC) ? 1.0f : 0.0f;

    // Batch-load the entire 16x64 A tile for this wave (16 x b64 per lane).
    v2f afrag[16];
    const float* abase = x + (size_t)rowc * D + hi * 2;
    #pragma unroll
    for (int kk = 0; kk < 16; ++kk) {
        afrag[kk].x = abase[kk * 4 + 0] * valid;
        afrag[kk].y = abase[kk * 4 + 1] * valid;
    }

    v8f acc[4] = {v8f{}, v8f{}, v8f{}, v8f{}};

    #pragma unroll
    for (int kk = 0; kk < 16; ++kk) {
        const int kb = kk * 4 + hi * 2;
        #pragma unroll
        for (int nb = 0; nb < 4; ++nb) {
            v2f b;
            b.x = ldsW[(kb + 0) * D + nb * 16 + mrow];
            b.y = ldsW[(kb + 1) * D + nb * 16 + mrow];
            acc[nb] = __builtin_amdgcn_wmma_f32_16x16x4_f32(
                /*neg_a=*/false, afrag[kk], /*neg_b=*/false, b,
                /*c_mod=*/(short)0, acc[nb],
                /*reuse_a=*/false, /*reuse_b=*/false);
        }
    }

    #pragma unroll
    for (int nb = 0; nb < 4; ++nb) {
        #pragma unroll
        for (int r = 0; r < 8; ++r) {
            int orow = row0 + r + hi * 8;
            if (orow < N_SRC) h[(size_t)orow * D + nb * 16 + mrow] = acc[nb][r];
        }
    }
}

// ---------------------------------------------------------------------------
// Kernel 3: a_src[i] = x[i]            . watt[0:64]
//           a_dst[j] = x[res_n_id[j]]  . watt[64:128]
// One wave32 per row; per-lane float2 loads + butterfly reduction.
// ---------------------------------------------------------------------------
__global__ __launch_bounds__(256) void gat_rowdot(const float* __restrict__ x,
                                                  const int* __restrict__ res_n_id,
                                                  const float* __restrict__ watt,
                                                  float* __restrict__ a_src,
                                                  float* __restrict__ a_dst) {
    int wid  = blockIdx.x * 8 + (threadIdx.x >> 5);
    int lane = threadIdx.x & 31;
    const float2* xp;
    const float2* wv;
    float* outp;
    if (wid < N_SRC) {
        xp = (const float2*)(x + (size_t)wid * D);
        wv = (const float2*)(watt);
        outp = a_src + wid;
    } else {
        int j = wid - N_SRC;
        if (j >= N_DST) return;  // wave-uniform exit
        xp = (const float2*)(x + (size_t)res_n_id[j] * D);
        wv = (const float2*)(watt + D);
        outp = a_dst + j;
    }
    float2 xv = xp[lane];
    float2 wvv = wv[lane];
    float s = xv.x * wvv.x + xv.y * wvv.y;
    #pragma unroll
    for (int off = 16; off > 0; off >>= 1) s += __shfl_xor(s, off, 32);
    if (lane == 0) *outp = s;
}

// ---------------------------------------------------------------------------
// Kernel 4: e = leaky_relu(a_src[es] + a_dst[ed]); segment max via
// order-preserving uint encoding + global_atomic_max_u32.
// ---------------------------------------------------------------------------
__global__ __launch_bounds__(256) void gat_edge_logit(const int* __restrict__ es,
                                                      const int* __restrict__ ed,
                                                      const float* __restrict__ a_src,
                                                      const float* __restrict__ a_dst,
                                                      float* __restrict__ e_buf,
                                                      unsigned* __restrict__ m_enc) {
    int k = blockIdx.x * blockDim.x + threadIdx.x;
    if (k >= E_CNT) return;
    int d = ed[k];
    float e = a_src[es[k]] + a_dst[d];
    e = (e > 0.0f) ? e : NEG_SLOPE * e;
    e_buf[k] = e;
    unsigned u = __float_as_uint(e);
    u = (u & 0x80000000u) ? ~u : (u | 0x80000000u);
    atomicMax(m_enc + d, u);
}

// ---------------------------------------------------------------------------
// Kernel 5: ex = exp(e - m[dst]); denom[dst] += ex (global_atomic_add_f32)
// ---------------------------------------------------------------------------
__global__ __launch_bounds__(256) void gat_edge_exp(const int* __restrict__ ed,
                                                    float* __restrict__ e_buf,
                                                    const unsigned* __restrict__ m_enc,
                                                    float* __restrict__ denom) {
    int k = blockIdx.x * blockDim.x + threadIdx.x;
    if (k >= E_CNT) return;
    int d = ed[k];
    unsigned u = m_enc[d];
    float m = (u & 0x80000000u) ? __uint_as_float(u & 0x7FFFFFFFu)
                                : __uint_as_float(~u);
    float ex = __expf(e_buf[k] - m);
    e_buf[k] = ex;  // overwrite logit with exp
    atomicAdd(denom + d, ex);
}

// ---------------------------------------------------------------------------
// Kernel 6: out[dst] += (ex/(denom[dst]+eps)) * h_src[src]
// One wave32 per edge: per-lane float2 gather (single global_load_b64, 256 B
// per wave) + 2 coalesced f32 atomics per lane (64 cols). h_src (25.6 MB) and
// out (5 MB) stay resident in the 192 MB L2, so atomics execute at L2 speed.
// ---------------------------------------------------------------------------
__global__ __launch_bounds__(256) void gat_scatter(const int* __restrict__ es,
                                                   const int* __restrict__ ed,
                                                   const float* __restrict__ e_buf,
                                                   const float* __restrict__ denom,
                                                   const float* __restrict__ h,
                                                   float* __restrict__ out) {
    int widx = blockIdx.x * 8 + (threadIdx.x >> 5);
    if (widx >= E_CNT) return;  // wave-uniform exit
    int lane = threadIdx.x & 31;
    int s = es[widx];
    int d = ed[widx];
    float alpha = e_buf[widx] / (denom[d] + 1e-16f);
    const float2* hp = (const float2*)(h + (size_t)s * D);
    float2 v = hp[lane];
    float* op = out + (size_t)d * D + lane * 2;
    atomicAdd(op + 0, alpha * v.x);
    atomicAdd(op + 1, alpha * v.y);
}

// ---------------------------------------------------------------------------
// Launch
// ---------------------------------------------------------------------------
extern "C" void kernel_launch(void* const* d_in, const int* in_sizes, int n_in,
                              void* d_out, int out_size, void* d_ws, size_t ws_size,
                              hipStream_t stream) {
    const float* x        = (const float*)d_in[0];
    const int*   res_n_id = (const int*)d_in[1];
    const int*   edge_src = (const int*)d_in[2];
    const int*   edge_dst = (const int*)d_in[3];
    const float* W        = (const float*)d_in[4];
    const float* att_src  = (const float*)d_in[5];
    const float* att_dst  = (const float*)d_in[6];
    const float* bias     = (const float*)d_in[7];
    float* out = (float*)d_out;

    char* ws = (char*)d_ws;
    float*    h_src = (float*)(ws);                    // 25,600,000 B
    float*    e_buf = (float*)(ws + 25600000);         //  5,000,000 B
    float*    a_src = (float*)(ws + 30600000);         //    400,000 B
    float*    a_dst = (float*)(ws + 31000000);         //     80,000 B
    float*    denom = (float*)(ws + 31080000);         //     80,000 B
    unsigned* m_enc = (unsigned*)(ws + 31160000);      //     80,000 B
    float*    watt  = (float*)(ws + 31240000);         //        512 B

    gat_init<<<(N_DST * D + 255) / 256, 256, 0, stream>>>(out, bias, denom, m_enc);
    gat_watt<<<1, 64, 0, stream>>>(W, att_src, att_dst, watt);
    gat_gemm_hsrc<<<(N_SRC + 63) / 64, 128, 0, stream>>>(x, W, h_src);
    gat_rowdot<<<(N_SRC + N_DST + 7) / 8, 256, 0, stream>>>(x, res_n_id, watt, a_src, a_dst);
    gat_edge_logit<<<(E_CNT + 255) / 256, 256, 0, stream>>>(edge_src, edge_dst, a_src,
                                                            a_dst, e_buf, m_enc);
    gat_edge_exp<<<(E_CNT + 255) / 256, 256, 0, stream>>>(edge_dst, e_buf, m_enc, denom);
    gat_scatter<<<(E_CNT + 7) / 8, 256, 0, stream>>>(edge_src, edge_dst, e_buf, denom,
                                                     h_src, out);
}